// HyperHeteroGNN_12678743458334
// MI455X (gfx1250) — compile-verified
//
#include <hip/hip_runtime.h>
#include <hip/hip_bf16.h>

typedef __attribute__((ext_vector_type(2))) float v2f;
typedef __attribute__((ext_vector_type(8))) float v8f;

#define HD 64  // hidden/feature dim

// ---------------------------------------------------------------------------
// Zero fill (grid-stride)
// ---------------------------------------------------------------------------
__global__ void zero_f32(float* __restrict__ p, long n) {
    long i = (long)blockIdx.x * blockDim.x + threadIdx.x;
    long stride = (long)gridDim.x * blockDim.x;
    for (; i < n; i += stride) p[i] = 0.0f;
}

// ---------------------------------------------------------------------------
// Node-feature GEMM: Y_i = X @ W_i for NW weight matrices (64x64 each).
// One wave32 computes a 16x64 output tile with V_WMMA_F32_16X16X4_F32.
// A (16x4 f32): v0 = K0 (lanes 0-15) / K2 (lanes 16-31), v1 = K1/K3.
// B (4x16 f32): v0 = row K0 (lanes 0-15) / K2 (lanes 16-31), v1 = K1/K3.
// C/D (16x16 f32): VGPR r = row r (lanes 0-15) / row r+8 (lanes 16-31).
// ---------------------------------------------------------------------------
template <int NW>
__global__ __launch_bounds__(256) void gemm_nodes(
    const float* __restrict__ X, int ntiles,
    const float* __restrict__ W0, const float* __restrict__ W1,
    const float* __restrict__ W2,
    float* __restrict__ Y0, float* __restrict__ Y1, float* __restrict__ Y2) {
    __shared__ float wlds[NW * HD * HD];

    const float* Ws[3] = {W0, W1, W2};
    for (int i = threadIdx.x; i < NW * HD * HD; i += 256)
        wlds[i] = Ws[i >> 12][i & 4095];
    __syncthreads();

    const int wave = threadIdx.x >> 5;
    const int tile = blockIdx.x * 8 + wave;
    if (tile >= ntiles) return;  // wave-uniform: EXEC stays all-ones for WMMA

    const int lane = threadIdx.x & 31;
    const int half = lane >> 4;   // 0: lanes 0-15, 1: lanes 16-31
    const int l16  = lane & 15;
    const int m0   = tile * 16;

    const float* xrow = X + (size_t)(m0 + l16) * HD + half * 2;

    v8f acc[NW][4];
#pragma unroll
    for (int i = 0; i < NW; ++i)
#pragma unroll
        for (int nt = 0; nt < 4; ++nt)
#pragma unroll
            for (int r = 0; r < 8; ++r) acc[i][nt][r] = 0.0f;

#pragma unroll
    for (int kk = 0; kk < 16; ++kk) {  // K = 64 in steps of 4
        v2f a = *(const v2f*)(xrow + kk * 4);
        const int kbase = (kk * 4 + half * 2) * HD;
#pragma unroll
        for (int i = 0; i < NW; ++i) {
            const float* wl = wlds + i * HD * HD + kbase;
#pragma unroll
            for (int nt = 0; nt < 4; ++nt) {
                v2f b;
                b[0] = wl[nt * 16 + l16];
                b[1] = wl[HD + nt * 16 + l16];
                acc[i][nt] = __builtin_amdgcn_wmma_f32_16x16x4_f32(
                    false, a, false, b, (short)0, acc[i][nt], false, false);
            }
        }
    }

    float* Ys[3] = {Y0, Y1, Y2};
#pragma unroll
    for (int i = 0; i < NW; ++i)
#pragma unroll
        for (int nt = 0; nt < 4; ++nt)
#pragma unroll
            for (int r = 0; r < 8; ++r)
                Ys[i][(size_t)(m0 + r + half * 8) * HD + nt * 16 + l16] =
                    acc[i][nt][r];
}

// ---------------------------------------------------------------------------
// Edge scatter: one wave per edge; gather 64 floats from msg[src], atomically
// add into acc[dst]; lane 0 bumps the float edge-count for the mean.
// ---------------------------------------------------------------------------
__global__ __launch_bounds__(256) void edge_scatter(
    const float* __restrict__ msg, const int* __restrict__ src,
    const int* __restrict__ dst, float* __restrict__ acc,
    float* __restrict__ cnt, int nE) {
    const int wave = (int)((blockIdx.x * (unsigned)blockDim.x + threadIdx.x) >> 5);
    const int lane = threadIdx.x & 31;
    if (wave >= nE) return;
    const int s = src[wave];
    const int d = dst[wave];
    const float m0 = msg[(size_t)s * HD + lane];
    const float m1 = msg[(size_t)s * HD + 32 + lane];
    atomicAdd(&acc[(size_t)d * HD + lane], m0);
    atomicAdd(&acc[(size_t)d * HD + 32 + lane], m1);
    if (lane == 0) atomicAdd(&cnt[d], 1.0f);
}

// ---------------------------------------------------------------------------
// Combine: relu(t_self + bias + acc0/max(c0,1) [+ acc1/max(c1,1)])
// ---------------------------------------------------------------------------
__global__ __launch_bounds__(256) void combine_bus(
    const float* __restrict__ tself, const float* __restrict__ bias,
    const float* __restrict__ acc0, const float* __restrict__ cnt0,
    const float* __restrict__ acc1, const float* __restrict__ cnt1,
    float* __restrict__ out, int n) {
    const long i = (long)blockIdx.x * blockDim.x + threadIdx.x;
    if (i >= (long)n * HD) return;
    const int node = (int)(i >> 6);
    const float c0 = fmaxf(cnt0[node], 1.0f);
    const float c1 = fmaxf(cnt1[node], 1.0f);
    const float v = tself[i] + bias[i & 63] + acc0[i] / c0 + acc1[i] / c1;
    out[i] = fmaxf(v, 0.0f);
}

__global__ __launch_bounds__(256) void combine_gen(
    const float* __restrict__ tself, const float* __restrict__ bias,
    const float* __restrict__ acc0, const float* __restrict__ cnt0,
    float* __restrict__ out, int n) {
    const long i = (long)blockIdx.x * blockDim.x + threadIdx.x;
    if (i >= (long)n * HD) return;
    const int node = (int)(i >> 6);
    const float c0 = fmaxf(cnt0[node], 1.0f);
    const float v = tself[i] + bias[i & 63] + acc0[i] / c0;
    out[i] = fmaxf(v, 0.0f);
}

// ---------------------------------------------------------------------------
// Sum-pool columns (64) over n_rows rows into g[0..63] (pre-zeroed).
// ---------------------------------------------------------------------------
__global__ __launch_bounds__(256) void pool_sum(const float* __restrict__ x,
                                                int n_rows,
                                                float* __restrict__ g) {
    __shared__ float lds[HD];
    if (threadIdx.x < HD) lds[threadIdx.x] = 0.0f;
    __syncthreads();
    const int col = threadIdx.x & 63;
    float s = 0.0f;
    for (int row = blockIdx.x * 4 + (threadIdx.x >> 6); row < n_rows;
         row += gridDim.x * 4)
        s += x[(size_t)row * HD + col];
    atomicAdd(&lds[col], s);
    __syncthreads();
    if (threadIdx.x < HD) atomicAdd(&g[threadIdx.x], lds[threadIdx.x]);
}

// ---------------------------------------------------------------------------
// Head MLP: out[16] = relu(g[128] @ W_h[128x64] + b_h) @ W_o[64x16] + b_o
// ---------------------------------------------------------------------------
__global__ __launch_bounds__(64) void head_mlp(
    const float* __restrict__ g, const float* __restrict__ W_h,
    const float* __restrict__ b_h, const float* __restrict__ W_o,
    const float* __restrict__ b_o, float* __restrict__ out) {
    __shared__ float h[HD];
    const int j = threadIdx.x;
    float s = b_h[j];
    for (int k = 0; k < 2 * HD; ++k) s += g[k] * W_h[k * HD + j];
    h[j] = fmaxf(s, 0.0f);
    __syncthreads();
    if (j < 16) {
        float o = b_o[j];
        for (int k = 0; k < HD; ++k) o += h[k] * W_o[k * 16 + j];
        out[j] = o;
    }
}

// ---------------------------------------------------------------------------
extern "C" void kernel_launch(void* const* d_in, const int* in_sizes, int n_in,
                              void* d_out, int out_size, void* d_ws,
                              size_t ws_size, hipStream_t stream) {
    (void)n_in; (void)out_size; (void)ws_size;

    const int NB = in_sizes[0] / HD;
    const int NG = in_sizes[1] / HD;
    const int EL = in_sizes[2];
    const int EX = in_sizes[4];

    const float* x_bus = (const float*)d_in[0];
    const float* x_gen = (const float*)d_in[1];
    const int* line_src = (const int*)d_in[2];
    const int* line_dst = (const int*)d_in[3];
    const int* g2b_src  = (const int*)d_in[4];
    const int* g2b_dst  = (const int*)d_in[5];
    const int* b2g_src  = (const int*)d_in[6];
    const int* b2g_dst  = (const int*)d_in[7];

    const float* Wsb[2]  = {(const float*)d_in[8],  (const float*)d_in[15]};
    const float* Wsg[2]  = {(const float*)d_in[9],  (const float*)d_in[16]};
    const float* Wl[2]   = {(const float*)d_in[10], (const float*)d_in[17]};
    const float* Wg2b[2] = {(const float*)d_in[11], (const float*)d_in[18]};
    const float* Wb2g[2] = {(const float*)d_in[12], (const float*)d_in[19]};
    const float* bsb[2]  = {(const float*)d_in[13], (const float*)d_in[20]};
    const float* bsg[2]  = {(const float*)d_in[14], (const float*)d_in[21]};
    const float* W_h = (const float*)d_in[22];
    const float* b_h = (const float*)d_in[23];
    const float* W_o = (const float*)d_in[24];
    const float* b_o = (const float*)d_in[25];

    // Workspace carve-up (fp32)
    const size_t NB64 = (size_t)NB * HD, NG64 = (size_t)NG * HD;
    float* p = (float*)d_ws;
    float* hb       = p; p += NB64;
    float* hg       = p; p += NG64;
    float* t_line   = p; p += NB64;
    float* t_b2g    = p; p += NB64;
    float* t_self_b = p; p += NB64;
    float* t_g2b    = p; p += NG64;
    float* t_self_g = p; p += NG64;
    float* acc_line = p; p += NB64;   // --- contiguous zero region starts here
    float* acc_g2b  = p; p += NB64;
    float* acc_b2g  = p; p += NG64;
    float* cnt_line = p; p += NB;
    float* cnt_g2b  = p; p += NB;
    float* cnt_b2g  = p; p += NG;     // --- contiguous zero region ends here
    float* gpool    = p; p += 2 * HD;
    const long zero_n = (long)(2 * NB64 + NG64 + 2 * NB + NG);

    const int btiles = NB / 16, gtiles = (NG + 15) / 16;
    const int bgemm_blocks = (btiles + 7) / 8;
    const int ggemm_blocks = (gtiles + 7) / 8;

    for (int l = 0; l < 2; ++l) {
        const float* hb_in = (l == 0) ? x_bus : hb;
        const float* hg_in = (l == 0) ? x_gen : hg;

        zero_f32<<<2048, 256, 0, stream>>>(acc_line, zero_n);

        gemm_nodes<3><<<bgemm_blocks, 256, 0, stream>>>(
            hb_in, btiles, Wl[l], Wb2g[l], Wsb[l], t_line, t_b2g, t_self_b);
        gemm_nodes<2><<<ggemm_blocks, 256, 0, stream>>>(
            hg_in, gtiles, Wg2b[l], Wsg[l], nullptr, t_g2b, t_self_g, nullptr);

        edge_scatter<<<(EL + 7) / 8, 256, 0, stream>>>(
            t_line, line_src, line_dst, acc_line, cnt_line, EL);
        edge_scatter<<<(EX + 7) / 8, 256, 0, stream>>>(
            t_g2b, g2b_src, g2b_dst, acc_g2b, cnt_g2b, EX);
        edge_scatter<<<(EX + 7) / 8, 256, 0, stream>>>(
            t_b2g, b2g_src, b2g_dst, acc_b2g, cnt_b2g, EX);

        combine_bus<<<(int)((NB64 + 255) / 256), 256, 0, stream>>>(
            t_self_b, bsb[l], acc_line, cnt_line, acc_g2b, cnt_g2b, hb, NB);
        combine_gen<<<(int)((NG64 + 255) / 256), 256, 0, stream>>>(
            t_self_g, bsg[l], acc_b2g, cnt_b2g, hg, NG);
    }

    zero_f32<<<1, 128, 0, stream>>>(gpool, 2 * HD);
    pool_sum<<<512, 256, 0, stream>>>(hb, NB, gpool);
    pool_sum<<<256, 256, 0, stream>>>(hg, NG, gpool + HD);
    head_mlp<<<1, 64, 0, stream>>>(gpool, W_h, b_h, W_o, b_o, (float*)d_out);
}